// DIGCN_batch_29454885716515
// MI455X (gfx1250) — compile-verified
//
#include <hip/hip_runtime.h>
#include <math.h>

typedef float v2f __attribute__((ext_vector_type(2)));
typedef float v8f __attribute__((ext_vector_type(8)));

#define D_IN  256
#define D_HID 128
#define D_EMB 128
#define D_OUT 16

// ---------------------------------------------------------------------------
// WT[c*K + k] = W[k*D + c]  (tiny; weights stay L2-resident afterwards)
// ---------------------------------------------------------------------------
__global__ void transpose_w_kernel(const float* __restrict__ in,
                                   float* __restrict__ out, int K, int D)
{
    const int idx = blockIdx.x * blockDim.x + threadIdx.x;
    if (idx < K * D) {
        const int k = idx / D;
        const int c = idx - k * D;
        out[c * K + k] = in[idx];
    }
}

// ---------------------------------------------------------------------------
// C[M,DOUT] = A[M,K] @ W[K,DOUT] (+ bias), f32 WMMA 16x16x4, W pre-transposed.
// One wave per 16x16 tile; tile-exit is wave-uniform so EXEC stays all-ones.
//
// A frag (16x4, 32b): lane l<16 -> A[rowBase+l][k..k+1], l>=16 -> [k+2..k+3]
//   = one aligned float2 load, pointer bumped +4 floats/step.
// B frag (4x16): VGPR0 = rows {k, k+2}, VGPR1 = {k+1, k+3} per half-wave
//   = one aligned float2 load from WT[col+lm][ka..ka+1], same +4 bump.
// ---------------------------------------------------------------------------
__global__ void gemm_wmma_f32_kernel(const float* __restrict__ A,
                                     const float* __restrict__ WT,
                                     float* __restrict__ C,
                                     const float* __restrict__ bias,
                                     int M, int K, int DOUTv,
                                     int totalTiles, int colTiles)
{
    const int gw = blockIdx.x * (blockDim.x >> 5) + (threadIdx.x >> 5);
    if (gw >= totalTiles) return;                  // uniform per wave

    const int rowTile = gw / colTiles;
    const int colTile = gw - rowTile * colTiles;
    const int lane = threadIdx.x & 31;
    const int half = lane >> 4;
    const int lm   = lane & 15;

    const int rowBase = rowTile * 16;
    const int colBase = colTile * 16;

    const float* pA = A  + (size_t)(rowBase + lm) * K + 2 * half;
    const float* pB = WT + (size_t)(colBase + lm) * K + 2 * half;

    v8f acc = {};
    const int steps = K >> 2;
    #pragma unroll 4
    for (int s = 0; s < steps; ++s) {
        v2f a = *(const v2f*)pA;  pA += 4;
        v2f b = *(const v2f*)pB;  pB += 4;
        // (neg_a, A, neg_b, B, c_mod, C, reuse_a, reuse_b)
        acc = __builtin_amdgcn_wmma_f32_16x16x4_f32(
                  false, a, false, b, (short)0, acc, false, false);
    }

    const float bv = bias ? bias[colBase + lm] : 0.0f;
    float* pC = C + (size_t)(rowBase + 8 * half) * DOUTv + colBase + lm;
    #pragma unroll
    for (int v = 0; v < 8; ++v) {
        const int r = rowBase + v + 8 * half;      // C/D layout: VGPR v -> M=v / v+8
        if (r < M) pC[(size_t)v * DOUTv] = acc[v] + bv;
    }
}

// ---------------------------------------------------------------------------
// agg[dst] += h[src] * w : one wave per edge, lane covers 4 of 128 features.
// 25.6MB agg target is L2-resident (192MB L2) -> atomics resolve in-cache.
// ---------------------------------------------------------------------------
__global__ void spmm_atomic_kernel(const float* __restrict__ H,
                                   float* __restrict__ AGG,
                                   const int* __restrict__ ei,   // [2,E]
                                   const float* __restrict__ ew,
                                   int E, int D)
{
    const int lane   = threadIdx.x & 31;
    const int wave   = (blockIdx.x * blockDim.x + threadIdx.x) >> 5;
    const int nwaves = (gridDim.x * blockDim.x) >> 5;

    for (int e = wave; e < E; e += nwaves) {
        const int   src = ei[e];
        const int   dst = ei[E + e];
        const float w   = ew[e];
        const float4 h4 = *(const float4*)(H + (size_t)src * D + lane * 4);
        float* p = AGG + (size_t)dst * D + lane * 4;
        atomicAdd(p + 0, h4.x * w);
        atomicAdd(p + 1, h4.y * w);
        atomicAdd(p + 2, h4.z * w);
        atomicAdd(p + 3, h4.w * w);
    }
}

// ---------------------------------------------------------------------------
__global__ void zero_kernel(float4* __restrict__ p, size_t n4)
{
    const size_t i = (size_t)blockIdx.x * blockDim.x + threadIdx.x;
    const size_t stride = (size_t)gridDim.x * blockDim.x;
    const float4 z = make_float4(0.f, 0.f, 0.f, 0.f);
    for (size_t j = i; j < n4; j += stride) p[j] = z;
}

// out = act(in + bias), float4-vectorized; D must be a power of two (128).
__global__ void bias_act_kernel(const float4* __restrict__ in,
                                const float* __restrict__ bias,
                                float4* __restrict__ out,
                                size_t total4, int D, int doRelu)
{
    const size_t i = (size_t)blockIdx.x * blockDim.x + threadIdx.x;
    const size_t stride = (size_t)gridDim.x * blockDim.x;
    const int mask4 = (D >> 2) - 1;
    for (size_t j = i; j < total4; j += stride) {
        const float4 b4 = *(const float4*)(bias + ((j & mask4) << 2));
        float4 v = in[j];
        v.x += b4.x; v.y += b4.y; v.z += b4.z; v.w += b4.w;
        if (doRelu) {
            v.x = fmaxf(v.x, 0.f); v.y = fmaxf(v.y, 0.f);
            v.z = fmaxf(v.z, 0.f); v.w = fmaxf(v.w, 0.f);
        }
        out[j] = v;
    }
}

// per-row log-softmax over 16 logits
__global__ void log_softmax16_kernel(const float* __restrict__ Y,
                                     float* __restrict__ out, int N)
{
    const int r = blockIdx.x * blockDim.x + threadIdx.x;
    if (r >= N) return;
    float v[D_OUT];
    float m = -INFINITY;
    #pragma unroll
    for (int j = 0; j < D_OUT; ++j) { v[j] = Y[(size_t)r * D_OUT + j]; m = fmaxf(m, v[j]); }
    float s = 0.0f;
    #pragma unroll
    for (int j = 0; j < D_OUT; ++j) s += __expf(v[j] - m);
    const float l = m + __logf(s);
    #pragma unroll
    for (int j = 0; j < D_OUT; ++j) out[(size_t)r * D_OUT + j] = v[j] - l;
}

// ---------------------------------------------------------------------------
extern "C" void kernel_launch(void* const* d_in, const int* in_sizes, int n_in,
                              void* d_out, int out_size, void* d_ws, size_t ws_size,
                              hipStream_t stream)
{
    const float* x    = (const float*)d_in[0];
    const float* ew   = (const float*)d_in[1];
    const float* W1   = (const float*)d_in[2];
    const float* b1   = (const float*)d_in[3];
    const float* W2   = (const float*)d_in[4];
    const float* b2   = (const float*)d_in[5];
    const float* W3   = (const float*)d_in[6];
    const float* b3   = (const float*)d_in[7];
    const float* fcw  = (const float*)d_in[8];
    const float* fcb  = (const float*)d_in[9];
    const int*   ei   = (const int*)d_in[10];

    const int N = in_sizes[0] / D_IN;   // 50000
    const int E = in_sizes[1];          // 625000

    // workspace layout
    float* B0   = (float*)d_ws;                       // [N,128] GEMM out
    float* B1   = B0 + (size_t)N * D_HID;             // [N,128] aggregation
    float* WT1  = B1 + (size_t)N * D_HID;             // [128,256]
    float* WT2  = WT1 + D_IN  * D_HID;                // [128,128]
    float* WT3  = WT2 + D_HID * D_HID;                // [128,128]
    float* WTfc = WT3 + D_HID * D_EMB;                // [16,128]
    float* Y    = B0;                                 // [N,16] reuse after L3

    float* outEmb = (float*)d_out;                    // [N,128]
    float* outLsm = outEmb + (size_t)N * D_EMB;       // [N,16]

    const int rowTiles = (N + 15) / 16;               // 3125
    const dim3 blk(256);

    auto transpose = [&](const float* W, float* WT, int K, int D) {
        const int tot = K * D;
        transpose_w_kernel<<<(tot + 255) / 256, blk, 0, stream>>>(W, WT, K, D);
    };
    auto gemm = [&](const float* A, const float* WT, float* C, const float* bias,
                    int K, int DOUTv) {
        const int colTiles = DOUTv / 16;
        const int total = rowTiles * colTiles;
        gemm_wmma_f32_kernel<<<(total + 7) / 8, blk, 0, stream>>>(
            A, WT, C, bias, N, K, DOUTv, total, colTiles);
    };
    auto zero = [&](float* p, size_t nElem) {
        zero_kernel<<<2048, blk, 0, stream>>>((float4*)p, nElem / 4);
    };
    auto spmm = [&](const float* H, float* AGG) {
        spmm_atomic_kernel<<<2048, blk, 0, stream>>>(H, AGG, ei, ew, E, D_HID);
    };
    auto biasAct = [&](const float* in, const float* b, float* out, int relu) {
        bias_act_kernel<<<2048, blk, 0, stream>>>((const float4*)in, b,
                                                  (float4*)out,
                                                  (size_t)N * D_HID / 4,
                                                  D_HID, relu);
    };

    // one-time (per launch) weight transposes
    transpose(W1, WT1, D_IN,  D_HID);
    transpose(W2, WT2, D_HID, D_HID);
    transpose(W3, WT3, D_HID, D_EMB);
    transpose(fcw, WTfc, D_EMB, D_OUT);

    // Layer 1: h = relu(spmm(x@W1) + b1)
    gemm(x, WT1, B0, nullptr, D_IN, D_HID);
    zero(B1, (size_t)N * D_HID);
    spmm(B0, B1);
    biasAct(B1, b1, B1, 1);

    // Layer 2
    gemm(B1, WT2, B0, nullptr, D_HID, D_HID);
    zero(B1, (size_t)N * D_HID);
    spmm(B0, B1);
    biasAct(B1, b2, B1, 1);

    // Layer 3 -> emb (no relu), straight into d_out
    gemm(B1, WT3, B0, nullptr, D_HID, D_EMB);
    zero(B1, (size_t)N * D_HID);
    spmm(B0, B1);
    biasAct(B1, b3, outEmb, 0);

    // y = emb @ fc_w + fc_b ; log-softmax
    gemm(outEmb, WTfc, Y, fcb, D_EMB, D_OUT);
    log_softmax16_kernel<<<(N + 255) / 256, blk, 0, stream>>>(Y, outLsm, N);
}